// SpatialOut_77781857730660
// MI455X (gfx1250) — compile-verified
//
#include <hip/hip_runtime.h>

typedef __attribute__((ext_vector_type(2))) float v2f;
typedef __attribute__((ext_vector_type(8))) float v8f;

#define NODE_DIM 128
#define HID 64

// ---------------------------------------------------------------------------
// Phase 0: zero the output and the num/den scratch (we accumulate atomically)
// ---------------------------------------------------------------------------
__global__ void zero_kernel(float* __restrict__ out, int nOut,
                            float* __restrict__ ws, int nWs) {
    int i = blockIdx.x * blockDim.x + threadIdx.x;
    if (i < nOut) out[i] = 0.0f;
    if (i < nWs)  ws[i]  = 0.0f;
}

// ---------------------------------------------------------------------------
// Phase 1: per-molecule mass-weighted sums (num[3B], den[B]) via run-coalesced
// atomics exploiting the sorted `batch` array. 4 atoms per thread.
// ---------------------------------------------------------------------------
__global__ void mass_centroid_accum(const float* __restrict__ pos,
                                    const int*   __restrict__ at_no,
                                    const int*   __restrict__ batch,
                                    const float* __restrict__ masses,
                                    float* __restrict__ num,
                                    float* __restrict__ den,
                                    int n) {
    const int CH = 4;
    int t = blockIdx.x * blockDim.x + threadIdx.x;
    int base = t * CH;
    if (base >= n) return;
    int end = base + CH; if (end > n) end = n;

    int   cur = batch[base];
    float sm = 0.0f, sx = 0.0f, sy = 0.0f, sz = 0.0f;
    for (int i = base; i < end; ++i) {
        int s = batch[i];
        if (s != cur) {
            atomicAdd(&num[3 * cur + 0], sx);
            atomicAdd(&num[3 * cur + 1], sy);
            atomicAdd(&num[3 * cur + 2], sz);
            atomicAdd(&den[cur], sm);
            sm = sx = sy = sz = 0.0f;
            cur = s;
        }
        float m = masses[at_no[i]];
        sx += m * pos[3 * i + 0];
        sy += m * pos[3 * i + 1];
        sz += m * pos[3 * i + 2];
        sm += m;
    }
    atomicAdd(&num[3 * cur + 0], sx);
    atomicAdd(&num[3 * cur + 1], sy);
    atomicAdd(&num[3 * cur + 2], sz);
    atomicAdd(&den[cur], sm);
}

__device__ __forceinline__ float silu_f(float x) {
    return x / (1.0f + __expf(-x));
}

// ---------------------------------------------------------------------------
// Phase 2: one wave32 per 16-atom tile.
//   H(16x64) = silu( X(16x128) @ W1(128x64) + b1 )  via v_wmma_f32_16x16x4_f32
//   scalar_out = H @ W2 + b2 ; out[batch] += scalar_out * |pos - centroid|^2
// A layout (16x4 f32): lane M = lane&15, VGPR0/1 hold K = kb, kb+1 (kb = 2*(lane>>4))
// B layout (4x16 f32): lane N = lane&15, VGPR0/1 hold K = kb, kb+1
// D layout (16x16 f32): lane N = lane&15, VGPR r holds M = r + 8*(lane>>4)
// ---------------------------------------------------------------------------
__global__ void mlp_spatial_kernel(const float* __restrict__ X,
                                   const float* __restrict__ pos,
                                   const int*   __restrict__ batch,
                                   const float* __restrict__ W1,
                                   const float* __restrict__ b1p,
                                   const float* __restrict__ W2,
                                   const float* __restrict__ b2p,
                                   const float* __restrict__ num,
                                   const float* __restrict__ den,
                                   float* __restrict__ out,
                                   int n, int nFullTiles) {
    int gid  = blockIdx.x * blockDim.x + threadIdx.x;
    int wave = gid >> 5;
    int lane = threadIdx.x & 31;
    int col  = lane & 15;
    int kb   = (lane >> 4) << 1;   // 0 or 2

    if (wave < nFullTiles) {
        int tileBase = wave * 16;

        // ---- spatial term for atom tileBase+col (both half-waves duplicate) ----
        int   atom = tileBase + col;
        int   seg  = batch[atom];
        float inv  = 1.0f / den[seg];
        float cx = num[3 * seg + 0] * inv;
        float cy = num[3 * seg + 1] * inv;
        float cz = num[3 * seg + 2] * inv;
        float dx = pos[3 * atom + 0] - cx;
        float dy = pos[3 * atom + 1] - cy;
        float dz = pos[3 * atom + 2] - cz;
        float sp = dx * dx + dy * dy + dz * dz;

        // ---- WMMA GEMM: 32 K-steps x 4 N-tiles ----
        const float* Xt = X + (size_t)(tileBase + col) * NODE_DIM + kb;
        v8f acc0 = {}, acc1 = {}, acc2 = {}, acc3 = {};
        for (int kk = 0; kk < NODE_DIM; kk += 4) {
            v2f a = *(const v2f*)(Xt + kk);               // 8B aligned
            const float* Wk = W1 + (size_t)(kk + kb) * HID + col;
            v2f b0, b1v, b2v, b3v;
            b0.x  = Wk[0];      b0.y  = Wk[HID + 0];
            b1v.x = Wk[16];     b1v.y = Wk[HID + 16];
            b2v.x = Wk[32];     b2v.y = Wk[HID + 32];
            b3v.x = Wk[48];     b3v.y = Wk[HID + 48];
            acc0 = __builtin_amdgcn_wmma_f32_16x16x4_f32(false, a, false, b0,  (short)0, acc0, false, false);
            acc1 = __builtin_amdgcn_wmma_f32_16x16x4_f32(false, a, false, b1v, (short)0, acc1, false, false);
            acc2 = __builtin_amdgcn_wmma_f32_16x16x4_f32(false, a, false, b2v, (short)0, acc2, false, false);
            acc3 = __builtin_amdgcn_wmma_f32_16x16x4_f32(false, a, false, b3v, (short)0, acc3, false, false);
        }

        // ---- bias + silu + dot with W2 (each lane owns 4 hidden columns) ----
        float bb0 = b1p[col], bb1 = b1p[16 + col], bb2 = b1p[32 + col], bb3 = b1p[48 + col];
        float w0  = W2[col],  w1  = W2[16 + col],  w2  = W2[32 + col],  w3  = W2[48 + col];
        float p[8];
        #pragma unroll
        for (int r = 0; r < 8; ++r) {
            float h0 = silu_f(acc0[r] + bb0);
            float h1 = silu_f(acc1[r] + bb1);
            float h2 = silu_f(acc2[r] + bb2);
            float h3 = silu_f(acc3[r] + bb3);
            p[r] = h0 * w0 + h1 * w1 + h2 * w2 + h3 * w3;
        }

        // ---- reduce over the 16 hidden-column lanes within each half-wave ----
        #pragma unroll
        for (int m = 1; m < 16; m <<= 1) {
            #pragma unroll
            for (int r = 0; r < 8; ++r) p[r] += __shfl_xor(p[r], m, 32);
        }

        // ---- one atomic per atom; lane 0 covers M=0..7, lane 16 covers M=8..15 ----
        float b2s = b2p[0];
        int srcbase = (lane & 16) ? 8 : 0;
        #pragma unroll
        for (int r = 0; r < 8; ++r) {
            float sp_r = __shfl(sp,  srcbase + r, 32);
            int   sg_r = __shfl(seg, srcbase + r, 32);
            if ((lane & 15) == 0) {
                atomicAdd(&out[sg_r], (p[r] + b2s) * sp_r);
            }
        }
    } else if (wave == nFullTiles) {
        // ---- scalar remainder path for n % 16 atoms (uniformly taken branch) ----
        int rem  = n - nFullTiles * 16;
        int atom = nFullTiles * 16 + lane;
        if (lane < rem) {
            const float* x = X + (size_t)atom * NODE_DIM;
            float pacc = 0.0f;
            for (int j = 0; j < HID; ++j) {
                float s = b1p[j];
                for (int k = 0; k < NODE_DIM; ++k) s += x[k] * W1[k * HID + j];
                pacc += silu_f(s) * W2[j];
            }
            int   seg = batch[atom];
            float inv = 1.0f / den[seg];
            float dx = pos[3 * atom + 0] - num[3 * seg + 0] * inv;
            float dy = pos[3 * atom + 1] - num[3 * seg + 1] * inv;
            float dz = pos[3 * atom + 2] - num[3 * seg + 2] * inv;
            atomicAdd(&out[seg], (pacc + b2p[0]) * (dx * dx + dy * dy + dz * dz));
        }
    }
}

// ---------------------------------------------------------------------------
extern "C" void kernel_launch(void* const* d_in, const int* in_sizes, int n_in,
                              void* d_out, int out_size, void* d_ws, size_t ws_size,
                              hipStream_t stream) {
    const float* x_scalar = (const float*)d_in[0];
    const float* pos      = (const float*)d_in[1];
    const int*   at_no    = (const int*)  d_in[2];
    const int*   batch    = (const int*)  d_in[3];
    const float* masses   = (const float*)d_in[4];
    const float* W1       = (const float*)d_in[5];
    const float* b1       = (const float*)d_in[6];
    const float* W2       = (const float*)d_in[7];
    const float* b2       = (const float*)d_in[8];
    float* out = (float*)d_out;

    int n = in_sizes[2];    // N atoms
    int B = out_size;       // number of molecules

    float* num = (float*)d_ws;                 // 3*B floats
    float* den = num + 3 * (size_t)B;          // B floats
    int nWs = 4 * B;

    {
        int total  = nWs > B ? nWs : B;
        int blocks = (total + 255) / 256;
        zero_kernel<<<blocks, 256, 0, stream>>>(out, B, num, nWs);
    }
    {
        int threads = (n + 3) / 4;
        int blocks  = (threads + 255) / 256;
        mass_centroid_accum<<<blocks, 256, 0, stream>>>(pos, at_no, batch, masses,
                                                        num, den, n);
    }
    {
        int nFull   = n / 16;
        int waves   = nFull + ((n & 15) ? 1 : 0);
        int threads = waves * 32;
        int blocks  = (threads + 255) / 256;
        mlp_spatial_kernel<<<blocks, 256, 0, stream>>>(x_scalar, pos, batch,
                                                       W1, b1, W2, b2,
                                                       num, den, out, n, nFull);
    }
}